// ParabolicPool1D_15161234555498
// MI455X (gfx1250) — compile-verified
//
#include <hip/hip_runtime.h>
#include <hip/hip_bf16.h>
#include <stdint.h>

// ---------------------------------------------------------------------------
// Max-plus parabolic dilation, stride-2.  B=8, C=64, L=65536 -> Lout=32768.
// Memory-bound: ~200 MB traffic -> ~8.6 us @ 23.3 TB/s.  WMMA inapplicable
// (max-plus, not linear algebra).  Uses the gfx1250 async copy engine
// (global_load_async_to_lds_b128 + s_wait_asynccnt) to stage tiles into LDS,
// conflict-free strided LDS reads, and non-temporal output stores so the
// 134 MB input can stay resident in the 192 MB L2 across replays.
// ---------------------------------------------------------------------------

#define L_IN      65536
#define L_OUT     32768
#define ROWS      512        // B*C
#define CCH       64
#define TILE_OUT  4096       // outputs per block
#define NLDS      8200       // LDS floats: 2*TILE_OUT inputs + halo, 16B-chunked
#define NCHUNK    2050       // NLDS/4 16-byte chunks
#define TPB       256

typedef int v4i __attribute__((ext_vector_type(4)));
typedef __attribute__((address_space(1))) v4i* gv4p;   // global AS pointer
typedef __attribute__((address_space(3))) v4i* lv4p;   // LDS AS pointer

__device__ __forceinline__ void async_copy16(const float* gsrc, float* ldst) {
#if defined(__HIP_DEVICE_COMPILE__)
# if __has_builtin(__builtin_amdgcn_global_load_async_to_lds_b128)
  __builtin_amdgcn_global_load_async_to_lds_b128(
      (gv4p)(v4i*)gsrc,      // reinterpret to v4i*, then addrspacecast to AS1
      (lv4p)(v4i*)ldst,      // reinterpret to v4i*, then addrspacecast to AS3
      0, 0);
# else
  unsigned loff = (unsigned)(unsigned long long)
                  (__attribute__((address_space(3))) char*)
                  (__attribute__((address_space(3))) void*)(void*)ldst;
  asm volatile("global_load_async_to_lds_b128 %0, %1, off"
               :: "v"(loff), "v"(gsrc) : "memory");
# endif
#else
  (void)gsrc; (void)ldst;
#endif
}

__device__ __forceinline__ void wait_async0() {
#if defined(__HIP_DEVICE_COMPILE__)
# if __has_builtin(__builtin_amdgcn_s_wait_asynccnt)
  __builtin_amdgcn_s_wait_asynccnt(0);
# else
  asm volatile("s_wait_asynccnt 0" ::: "memory");
# endif
#endif
}

__global__ __launch_bounds__(TPB) void ParabolicPool1D_kernel(
    const float* __restrict__ f, const float* __restrict__ t,
    float* __restrict__ out) {
  __shared__ float lds[NLDS];

  const int tid = threadIdx.x;
  const int row = blockIdx.y;                 // b*64 + c
  const int ch  = row & (CCH - 1);
  const int ob  = blockIdx.x * TILE_OUT;      // first output of this tile
  const int A   = 2 * ob - 4;                 // row-local input idx of lds[0]
  const int rowbase = row * L_IN;
  const int TOTAL   = ROWS * L_IN;

  // ---- stage tile into LDS via gfx1250 async copy (ASYNCcnt path) ----
  // 256 lanes x 16B = 4 KB per round; 8 full rounds + 2 leftover chunks.
  #pragma unroll
  for (int r = 0; r < 8; ++r) {
    const int chunk = r * TPB + tid;
    int g = rowbase + A + chunk * 4;          // 16B-aligned global float index
    g = g < 0 ? 0 : (g > TOTAL - 4 ? TOTAL - 4 : g);   // memory-safety clamp
    async_copy16(f + g, &lds[chunk * 4]);
  }
  if (tid < NCHUNK - 8 * TPB) {               // chunks 2048, 2049
    const int chunk = 8 * TPB + tid;
    int g = rowbase + A + chunk * 4;
    g = g < 0 ? 0 : (g > TOTAL - 4 ? TOTAL - 4 : g);
    async_copy16(f + g, &lds[chunk * 4]);
  }
  wait_async0();          // this wave's async copies complete
  __syncthreads();        // make all waves' LDS writes visible

  // ---- patch row edges: out-of-row taps must be -inf (only 2 blocks/row) --
  if (A < 0) {                                // first tile: idx 0..3 -> x<0
    if (tid < 4) lds[tid] = -INFINITY;
  }
  if (A + NLDS > L_IN) {                      // last tile: idx 8196..8199 -> x>=L
    if (tid < 4) lds[NLDS - 4 + tid] = -INFINITY;
  }
  __syncthreads();

  // ---- compute: symmetric parabola weights {0, inv, 4inv, 9inv} ----------
  const float inv = -0.25f / t[ch];           // uniform per block -> SGPR
  const float h1 = inv, h2 = 4.0f * inv, h3 = 9.0f * inv;
  const float2* l2 = (const float2*)lds;
  float* __restrict__ orow = out + row * L_OUT + ob;

  // Strided output mapping X = tid + 256k:
  //  - LDS reads at byte 8*X per lane -> 32 lanes cover all 64 banks, no conflict
  //  - stores are 128B-contiguous per wave.
  #pragma unroll
  for (int k = 0; k < 16; ++k) {
    const int X = tid + k * TPB;
    const float2 a = l2[X];                   // floats 2X   (unused), 2X+1 (d=-3)
    const float2 b = l2[X + 1];               // 2X+2 (d=-2), 2X+3 (d=-1)
    const float2 c = l2[X + 2];               // 2X+4 (d= 0), 2X+5 (d=+1)
    const float2 d = l2[X + 3];               // 2X+6 (d=+2), 2X+7 (d=+3)

    const float m1 = fmaxf(b.y, c.y) + h1;    // |d| = 1
    const float m2 = fmaxf(b.x, d.x) + h2;    // |d| = 2
    const float m3 = fmaxf(a.y, d.y) + h3;    // |d| = 3
    float v = fmaxf(fmaxf(c.x, m1), fmaxf(m2, m3));
    v = fmaxf(v, 0.0f);                       // running max starts at 0

    __builtin_nontemporal_store(v, orow + X); // keep f resident in L2
  }
}

extern "C" void kernel_launch(void* const* d_in, const int* in_sizes, int n_in,
                              void* d_out, int out_size, void* d_ws, size_t ws_size,
                              hipStream_t stream) {
  (void)in_sizes; (void)n_in; (void)out_size; (void)d_ws; (void)ws_size;
  const float* f = (const float*)d_in[0];     // (8, 64, 65536) f32
  const float* t = (const float*)d_in[1];     // (64,) f32
  float* out = (float*)d_out;                 // (8, 64, 32768) f32

  dim3 grid(L_OUT / TILE_OUT, ROWS);          // (8, 512)
  ParabolicPool1D_kernel<<<grid, TPB, 0, stream>>>(f, t, out);
}